// GraphMatchingNetwork_15126874816762
// MI455X (gfx1250) — compile-verified
//
#include <hip/hip_runtime.h>
#include <hip/hip_bf16.h>
#include <math.h>

typedef __attribute__((ext_vector_type(2))) float v2f;
typedef __attribute__((ext_vector_type(8))) float v8f;

#define WMMA4(a, b, c) \
  __builtin_amdgcn_wmma_f32_16x16x4_f32(false, (a), false, (b), (short)0, (c), false, false)

// ---------------------------------------------------------------------------
// Generic GEMM: C[M,N] = act(A[M,K] @ B[K,N] + bias), optional batch strides.
// One wave (32 lanes) computes one 16x16 C tile via V_WMMA_F32_16X16X4_F32.
// aligned!=0 => M,N,K multiples of 16, lda/ldb multiples of 4: b128 loads,
// no bounds checks (fast path used by every GEMM except the tiny readout).
// ---------------------------------------------------------------------------
__global__ __launch_bounds__(32)
void gemm_f32_wmma(const float* __restrict__ A, const float* __restrict__ B,
                   const float* __restrict__ bias, float* __restrict__ C,
                   int M, int N, int K, int lda, int ldb, int ldc,
                   long long sA, long long sB, long long sC, int relu, int aligned)
{
  const int bz = blockIdx.z;
  A += (long long)bz * sA;
  B += (long long)bz * sB;
  C += (long long)bz * sC;
  const int lane = threadIdx.x;
  const int row0 = blockIdx.x * 16, col0 = blockIdx.y * 16;

  __shared__ float As[16][20];   // row stride 80B -> float4 stores stay 16B aligned
  __shared__ float Bs[16][20];

  v8f acc = {0.f, 0.f, 0.f, 0.f, 0.f, 0.f, 0.f, 0.f};
  const int arow = lane & 15;        // A row / B col within tile
  const int koff = (lane >> 4) * 2;  // K sub-offset per half-wave

  for (int k0 = 0; k0 < K; k0 += 16) {
    __syncthreads();
    if (aligned) {
#pragma unroll
      for (int e = 0; e < 2; ++e) {
        int idx = e * 32 + lane;      // 0..63 float4 slots of the 16x16 tile
        int r = idx >> 2, c4 = idx & 3;
        *(float4*)&As[r][c4 * 4] =
            *(const float4*)(A + (long long)(row0 + r) * lda + k0 + c4 * 4);
        *(float4*)&Bs[r][c4 * 4] =
            *(const float4*)(B + (long long)(k0 + r) * ldb + col0 + c4 * 4);
      }
    } else {
#pragma unroll
      for (int e = 0; e < 8; ++e) {
        int idx = e * 32 + lane;      // 0..255
        int r = idx >> 4, c = idx & 15;
        int gr = row0 + r, gc = k0 + c;
        As[r][c] = (gr < M && gc < K) ? A[(long long)gr * lda + gc] : 0.f;
        int br = k0 + r, bc = col0 + c;
        Bs[r][c] = (br < K && bc < N) ? B[(long long)br * ldb + bc] : 0.f;
      }
    }
    __syncthreads();
#pragma unroll
    for (int kk = 0; kk < 16; kk += 4) {
      v2f a, b;
      a.x = As[arow][kk + koff];
      a.y = As[arow][kk + koff + 1];
      b.x = Bs[kk + koff][arow];
      b.y = Bs[kk + koff + 1][arow];
      acc = WMMA4(a, b, acc);
    }
  }

  const int col = col0 + arow;
  if (aligned || col < N) {
    const float bv = bias ? bias[col] : 0.f;
#pragma unroll
    for (int v = 0; v < 8; ++v) {
      int r = row0 + v + (lane >> 4) * 8;
      if (aligned || r < M) {
        float x = acc[v] + bv;
        if (relu) x = fmaxf(x, 0.f);
        C[(long long)r * ldc + col] = x;
      }
    }
  }
}

// ---------------------------------------------------------------------------
// Fused attention with online softmax (flash style), H=128 fixed.
// O[b, qt*16+..] = softmax(Q K^T * scale + keymask) V
// keymask[b, j] == 0  =>  key j masked (score -> -3e38).
// One wave per (batch, 16-query tile). Nk must be a multiple of 16.
// ---------------------------------------------------------------------------
__global__ __launch_bounds__(32)
void attn_f32_wmma(const float* __restrict__ Q, const float* __restrict__ K,
                   const float* __restrict__ V, const float* __restrict__ mask,
                   float* __restrict__ O, int Nk,
                   long long sQO, long long sKV, long long sM, float scale)
{
  const int b = blockIdx.y;
  const int qt = blockIdx.x;
  const float* Qb = Q + (long long)b * sQO + (long long)qt * 16 * 128;
  const float* Kb = K + (long long)b * sKV;
  const float* Vb = V + (long long)b * sKV;
  const float* mb = mask + (long long)b * sM;
  float* Ob = O + (long long)b * sQO + (long long)qt * 16 * 128;

  __shared__ float Qs[16][132];   // row stride 528B -> 16B aligned float4 slots
  __shared__ float Ks[16][132];
  __shared__ float Vs[16][132];
  __shared__ float Ps[16][17];

  const int lane = threadIdx.x;
  const int arow = lane & 15;
  const int half = lane >> 4;
  const int koff = half * 2;

  // Load 16x128 Q tile (512 float4s, 16 per lane)
  {
    const float4* Qg = (const float4*)Qb;
#pragma unroll
    for (int e = 0; e < 16; ++e) {
      int idx = e * 32 + lane;
      *(float4*)&Qs[idx >> 5][(idx & 31) * 4] = Qg[idx];
    }
  }

  v8f acc[8];
#pragma unroll
  for (int cb = 0; cb < 8; ++cb) acc[cb] = (v8f){0.f,0.f,0.f,0.f,0.f,0.f,0.f,0.f};
  float mrow[8], lrow[8];
#pragma unroll
  for (int v = 0; v < 8; ++v) { mrow[v] = -3.0e38f; lrow[v] = 0.f; }

  for (int kc = 0; kc < Nk; kc += 16) {
    __syncthreads();
    {
      const float4* Kg = (const float4*)(Kb + (long long)kc * 128);
      const float4* Vg = (const float4*)(Vb + (long long)kc * 128);
#pragma unroll
      for (int e = 0; e < 16; ++e) {
        int idx = e * 32 + lane;
        int r = idx >> 5, c = (idx & 31) * 4;
        *(float4*)&Ks[r][c] = Kg[idx];
        *(float4*)&Vs[r][c] = Vg[idx];
      }
    }
    // Prefetch next K/V chunk (gfx1250 global_prefetch_b8) while this chunk
    // feeds 64 WMMAs below; 32 lanes x 256B stride covers the 8KB chunk.
    if (kc + 16 < Nk) {
      const float* nK = Kb + (long long)(kc + 16) * 128;
      const float* nV = Vb + (long long)(kc + 16) * 128;
      __builtin_prefetch(nK + lane * 64, 0, 3);
      __builtin_prefetch(nV + lane * 64, 0, 3);
    }
    __syncthreads();

    // s = Q @ K^T  (contraction over H=128)
    v8f s = {0.f,0.f,0.f,0.f,0.f,0.f,0.f,0.f};
#pragma unroll
    for (int kk = 0; kk < 128; kk += 4) {
      v2f a, bf;
      a.x = Qs[arow][kk + koff];
      a.y = Qs[arow][kk + koff + 1];
      bf.x = Ks[arow][kk + koff];      // B[k][col] = K[col][k], col == arow
      bf.y = Ks[arow][kk + koff + 1];
      s = WMMA4(a, bf, s);
    }

    // scale + key mask (per column = key kc+arow)
    const bool keep = (mb[kc + arow] != 0.f);
    float sv[8];
#pragma unroll
    for (int v = 0; v < 8; ++v) sv[v] = keep ? (s[v] * scale) : -3.0e38f;

    // row-max across the 16 lanes of this half-wave
    float cm[8];
#pragma unroll
    for (int v = 0; v < 8; ++v) cm[v] = sv[v];
    for (int off = 1; off < 16; off <<= 1) {
#pragma unroll
      for (int v = 0; v < 8; ++v) cm[v] = fmaxf(cm[v], __shfl_xor(cm[v], off, 32));
    }

    float sf[8], p[8], ls[8];
#pragma unroll
    for (int v = 0; v < 8; ++v) {
      float mn = fmaxf(mrow[v], cm[v]);
      sf[v] = __expf(mrow[v] - mn);
      mrow[v] = mn;
      p[v] = __expf(sv[v] - mn);
      ls[v] = p[v];
    }
    for (int off = 1; off < 16; off <<= 1) {
#pragma unroll
      for (int v = 0; v < 8; ++v) ls[v] += __shfl_xor(ls[v], off, 32);
    }
#pragma unroll
    for (int v = 0; v < 8; ++v) lrow[v] = lrow[v] * sf[v] + ls[v];
#pragma unroll
    for (int cb = 0; cb < 8; ++cb)
#pragma unroll
      for (int v = 0; v < 8; ++v) acc[cb][v] *= sf[v];

    // P tile (C layout) -> LDS so it can re-enter as the A operand
#pragma unroll
    for (int v = 0; v < 8; ++v) Ps[v + half * 8][arow] = p[v];
    __syncthreads();

    // acc += P @ V_chunk  (contraction over 16 keys)
#pragma unroll
    for (int cb = 0; cb < 8; ++cb) {
#pragma unroll
      for (int kk = 0; kk < 16; kk += 4) {
        v2f a, bf;
        a.x = Ps[arow][kk + koff];
        a.y = Ps[arow][kk + koff + 1];
        bf.x = Vs[kk + koff][cb * 16 + arow];
        bf.y = Vs[kk + koff + 1][cb * 16 + arow];
        acc[cb] = WMMA4(a, bf, acc[cb]);
      }
    }
  }

#pragma unroll
  for (int cb = 0; cb < 8; ++cb)
#pragma unroll
    for (int v = 0; v < 8; ++v) {
      int r = v + half * 8;
      Ob[(long long)r * 128 + cb * 16 + arow] = acc[cb][v] / lrow[v];
    }
}

// ---------------------------------------------------------------------------
// Small VALU kernels
// ---------------------------------------------------------------------------
// agg_e[b,i,k] = sum_j adj[b,i,j] * E[b,i,j,k]
__global__ void agg_edge_k(const float* __restrict__ adj, const float* __restrict__ E,
                           float* __restrict__ outp, int N, int ein)
{
  int bi = blockIdx.x;           // b*N + i
  int b = bi / N, i = bi % N;
  int k = threadIdx.x;
  if (k >= ein) return;
  const float* ar = adj + ((long long)b * N + i) * N;
  const float* Er = E + (((long long)b * N + i) * N) * ein;
  float s = 0.f;
  for (int j = 0; j < N; ++j) s += ar[j] * Er[(long long)j * ein + k];
  outp[(long long)bi * ein + k] = s;
}

// inc[b,i,k] = sum_j adj[b,i,j]*(cE[b,i,j,k] + cE[b,j,i,k])
__global__ void inc_edge_k(const float* __restrict__ adj, const float* __restrict__ cE,
                           float* __restrict__ outp, int N)
{
  int bi = blockIdx.x;
  int b = bi / N, i = bi % N;
  int k = threadIdx.x;  // 128
  const float* ab = adj + (long long)b * N * N;
  const float* cb = cE + (long long)b * N * N * 128;
  float s = 0.f;
  for (int j = 0; j < N; ++j) {
    float a = ab[i * N + j];
    s += a * cb[((long long)i * N + j) * 128 + k];
    s += a * cb[((long long)j * N + i) * 128 + k];
  }
  outp[(long long)bi * 128 + k] = s;
}

__global__ void relu_add2_k(const float* a, const float* b, float* o, int n)
{
  int i = blockIdx.x * blockDim.x + threadIdx.x;
  if (i < n) o[i] = fmaxf(a[i] + b[i], 0.f);
}

__global__ void add3_k(float* h, const float* a, const float* b, int n)
{
  int i = blockIdx.x * blockDim.x + threadIdx.x;
  if (i < n) h[i] = h[i] + a[i] + b[i];
}

// E_out[b,i,j,k] = relu(Pi[b,i,k] + Pj[b,j,k] + Ppe[b,i,j,k] + blu[k]) * adj[b,i,j]
__global__ void eout_combine_k(const float* __restrict__ Pi, const float* __restrict__ Pj,
                               const float* __restrict__ Ppe, const float* __restrict__ blu,
                               const float* __restrict__ adj, float* __restrict__ Eout, int N)
{
  long long bij = blockIdx.x;
  int k = threadIdx.x;  // 128
  int b = (int)(bij / (N * N));
  int rem = (int)(bij % (N * N));
  int i = rem / N, j = rem % N;
  float v = Pi[((long long)b * N + i) * 128 + k] + Pj[((long long)b * N + j) * 128 + k] +
            Ppe[bij * 128 + k] + blu[k];
  Eout[bij * 128 + k] = fmaxf(v, 0.f) * adj[(long long)b * N * N + rem];
}

__global__ void i2f_k(const int* a, float* b, int n)
{
  int i = blockIdx.x * blockDim.x + threadIdx.x;
  if (i < n) b[i] = (float)a[i];
}

// g[b, goff+k] = sum_i h[b,i,k]*m[b,i] / (sum_i m[b,i] + 1e-9)
__global__ void readout_g_k(const float* __restrict__ h, const float* __restrict__ mf,
                            float* __restrict__ g, int N, int goff)
{
  int b = blockIdx.x;
  int k = threadIdx.x;  // 128
  float ms = 0.f, s = 0.f;
  for (int i = 0; i < N; ++i) {
    float m = mf[b * N + i];
    ms += m;
    s += m * h[((long long)b * N + i) * 128 + k];
  }
  g[(long long)b * 256 + goff + k] = s / (ms + 1e-9f);
}

// ---------------------------------------------------------------------------
// Host orchestration
// ---------------------------------------------------------------------------
static inline void gemm(hipStream_t st, const float* A, const float* B, const float* bias,
                        float* C, int M, int N, int K, int lda, int ldb, int ldc,
                        long long sA, long long sB, long long sC, int relu, int batch)
{
  const int aligned = ((M % 16) == 0) && ((N % 16) == 0) && ((K % 16) == 0) &&
                      ((lda % 4) == 0) && ((ldb % 4) == 0);
  dim3 g((M + 15) / 16, (N + 15) / 16, batch);
  hipLaunchKernelGGL(gemm_f32_wmma, g, dim3(32), 0, st,
                     A, B, bias, C, M, N, K, lda, ldb, ldc, sA, sB, sC, relu, aligned);
}

extern "C" void kernel_launch(void* const* d_in, const int* in_sizes, int n_in,
                              void* d_out, int out_size, void* d_ws, size_t ws_size,
                              hipStream_t stream)
{
  if (n_in < 38) return;
  constexpr int Bd = 4, Nd = 48, Hd = 128;
  constexpr int HB = Bd * Nd * Hd;            // 24576
  constexpr int EB = Bd * Nd * Nd * Hd;       // 1179648
  const float SCALE = 0.088388347648318447f;  // 1/sqrt(128)

  const float* x1   = (const float*)d_in[0];
  const float* adj1 = (const float*)d_in[1];
  const float* e1   = (const float*)d_in[2];
  const float* x2   = (const float*)d_in[3];
  const float* adj2 = (const float*)d_in[4];
  const float* e2   = (const float*)d_in[5];
  const int*   m1i  = (const int*)d_in[6];
  const int*   m2i  = (const int*)d_in[7];

  // Param index maps: jax pytree (sorted keys) vs dict-insertion order.
  int lnw[2], lnb[2], lew[2], leb[2], luw[2], lub[2];
  int cnqw, cnqb, cnkw, cnkb, cnvw, cnvb;
  int ceqw, ceqb, cekw, cekb, cevw, cevb;
  int e2nw, e2nb, ro1w, ro1b, ro2w, ro2b;
  if (in_sizes[8] == 64 * 128) {
    // insertion: layers[(ln,le,lu)x2], cn(q,k,v), ce(q,k,v), e2n, ro1, ro2
    int i = 8;
    for (int l = 0; l < 2; ++l) { lnw[l]=i++; lnb[l]=i++; lew[l]=i++; leb[l]=i++; luw[l]=i++; lub[l]=i++; }
    cnqw=i++; cnqb=i++; cnkw=i++; cnkb=i++; cnvw=i++; cnvb=i++;
    ceqw=i++; ceqb=i++; cekw=i++; cekb=i++; cevw=i++; cevb=i++;
    e2nw=i++; e2nb=i++; ro1w=i++; ro1b=i++; ro2w=i++; ro2b=i++;
  } else {
    // sorted: ce(k,q,v), cn(k,q,v), e2n, layers[(le,ln,lu)x2], ro1, ro2
    int i = 8;
    cekw=i++; cekb=i++; ceqw=i++; ceqb=i++; cevw=i++; cevb=i++;
    cnkw=i++; cnkb=i++; cnqw=i++; cnqb=i++; cnvw=i++; cnvb=i++;
    e2nw=i++; e2nb=i++;
    for (int l = 0; l < 2; ++l) { lew[l]=i++; leb[l]=i++; lnw[l]=i++; lnb[l]=i++; luw[l]=i++; lub[l]=i++; }
    ro1w=i++; ro1b=i++; ro2w=i++; ro2b=i++;
  }
  auto P = [&](int i) { return (const float*)d_in[i]; };

  // Workspace carve-out
  float* wp = (float*)d_ws;
  auto alloc = [&](long long n) { float* p = wp; wp += n; return p; };
  float* hb1 = alloc(HB); float* hb2 = alloc(HB);
  float* t0 = alloc(HB); float* t1 = alloc(HB); float* t2 = alloc(HB);
  float* Pi = alloc(HB); float* Pj = alloc(HB);
  float* ch1 = alloc(HB); float* ch2 = alloc(HB);
  float* Qn1 = alloc(HB); float* Kn1 = alloc(HB); float* Vn1 = alloc(HB);
  float* Qn2 = alloc(HB); float* Kn2 = alloc(HB); float* Vn2 = alloc(HB);
  float* gbuf = alloc(Bd * 256); float* r1 = alloc(Bd * 128);
  float* mf1 = alloc(Bd * Nd); float* mf2 = alloc(Bd * Nd);
  float* Eb1a = alloc(EB); float* Eb1b = alloc(EB);
  float* Eb2a = alloc(EB); float* Eb2b = alloc(EB);
  float* Esc = alloc(EB); float* Ppe = alloc(EB);
  float* QE1 = alloc(EB); float* KE1 = alloc(EB); float* VE1 = alloc(EB);
  float* QE2 = alloc(EB); float* KE2 = alloc(EB); float* VE2 = alloc(EB);
  float* cE1 = alloc(EB); float* cE2 = alloc(EB);
  (void)ws_size;

  hipLaunchKernelGGL(i2f_k, dim3(1), dim3(256), 0, stream, m1i, mf1, Bd * Nd);
  hipLaunchKernelGGL(i2f_k, dim3(1), dim3(256), 0, stream, m2i, mf2, Bd * Nd);

  for (int l = 0; l < 2; ++l) {
    const int din = (l == 0) ? 64 : 128;
    const int ein = (l == 0) ? 32 : 128;
    const float* hin[2] = { l == 0 ? x1 : hb1, l == 0 ? x2 : hb2 };
    const float* Ein[2] = { l == 0 ? e1 : Eb1a, l == 0 ? e2 : Eb2a };
    float* hbo[2] = { hb1, hb2 };
    float* Eout[2] = { l == 0 ? Eb1a : Eb1b, l == 0 ? Eb2a : Eb2b };
    const float* adj[2] = { adj1, adj2 };

    // ---- conv for both graphs ----
    for (int gph = 0; gph < 2; ++gph) {
      // agg_n = adj @ h  : batched [48x48]@[48xdin]
      gemm(stream, adj[gph], hin[gph], nullptr, t0, Nd, din, Nd, Nd, din, din,
           (long long)Nd * Nd, (long long)Nd * din, (long long)Nd * din, 0, Bd);
      // hn = relu(agg_n @ Wln + bln)
      gemm(stream, t0, P(lnw[l]), P(lnb[l]), t1, Bd * Nd, Hd, din, din, Hd, Hd, 0, 0, 0, 1, 1);
      // agg_e and he = relu(agg_e @ Wle + ble)
      hipLaunchKernelGGL(agg_edge_k, dim3(Bd * Nd), dim3(128), 0, stream, adj[gph], Ein[gph], t0, Nd, ein);
      gemm(stream, t0, P(lew[l]), P(leb[l]), t2, Bd * Nd, Hd, ein, ein, Hd, Hd, 0, 0, 0, 1, 1);
      // h_out = relu(hn + he)
      hipLaunchKernelGGL(relu_add2_k, dim3((HB + 255) / 256), dim3(256), 0, stream, t1, t2, hbo[gph], HB);
      // e_proj = relu(E @ Wle + ble)
      gemm(stream, Ein[gph], P(lew[l]), P(leb[l]), Esc, Bd * Nd * Nd, Hd, ein, ein, Hd, Hd, 0, 0, 0, 1, 1);
      // partial lu projections: Pi = h_out@Wlu[0:H], Pj = h_out@Wlu[H:2H], Ppe = e_proj@Wlu[2H:3H]
      gemm(stream, hbo[gph], P(luw[l]), nullptr, Pi, Bd * Nd, Hd, Hd, Hd, Hd, Hd, 0, 0, 0, 0, 1);
      gemm(stream, hbo[gph], P(luw[l]) + Hd * Hd, nullptr, Pj, Bd * Nd, Hd, Hd, Hd, Hd, Hd, 0, 0, 0, 0, 1);
      gemm(stream, Esc, P(luw[l]) + 2 * Hd * Hd, nullptr, Ppe, Bd * Nd * Nd, Hd, Hd, Hd, Hd, Hd, 0, 0, 0, 0, 1);
      hipLaunchKernelGGL(eout_combine_k, dim3(Bd * Nd * Nd), dim3(128), 0, stream,
                         Pi, Pj, Ppe, P(lub[l]), adj[gph], Eout[gph], Nd);
    }

    // ---- cross node attention ----
    gemm(stream, hb1, P(cnqw), P(cnqb), Qn1, Bd * Nd, Hd, Hd, Hd, Hd, Hd, 0, 0, 0, 0, 1);
    gemm(stream, hb1, P(cnkw), P(cnkb), Kn1, Bd * Nd, Hd, Hd, Hd, Hd, Hd, 0, 0, 0, 0, 1);
    gemm(stream, hb1, P(cnvw), P(cnvb), Vn1, Bd * Nd, Hd, Hd, Hd, Hd, Hd, 0, 0, 0, 0, 1);
    gemm(stream, hb2, P(cnqw), P(cnqb), Qn2, Bd * Nd, Hd, Hd, Hd, Hd, Hd, 0, 0, 0, 0, 1);
    gemm(stream, hb2, P(cnkw), P(cnkb), Kn2, Bd * Nd, Hd, Hd, Hd, Hd, Hd, 0, 0, 0, 0, 1);
    gemm(stream, hb2, P(cnvw), P(cnvb), Vn2, Bd * Nd, Hd, Hd, Hd, Hd, Hd, 0, 0, 0, 0, 1);
    hipLaunchKernelGGL(attn_f32_wmma, dim3(Nd / 16, Bd), dim3(32), 0, stream,
                       Qn1, Kn2, Vn2, mf2, ch1, Nd, (long long)Nd * Hd, (long long)Nd * Hd, Nd, SCALE);
    hipLaunchKernelGGL(attn_f32_wmma, dim3(Nd / 16, Bd), dim3(32), 0, stream,
                       Qn2, Kn1, Vn1, mf1, ch2, Nd, (long long)Nd * Hd, (long long)Nd * Hd, Nd, SCALE);

    // ---- cross edge attention (fused flash, keys masked by adj of key graph) ----
    const int NE = Nd * Nd;  // 2304
    gemm(stream, Eout[0], P(ceqw), P(ceqb), QE1, Bd * NE, Hd, Hd, Hd, Hd, Hd, 0, 0, 0, 0, 1);
    gemm(stream, Eout[0], P(cekw), P(cekb), KE1, Bd * NE, Hd, Hd, Hd, Hd, Hd, 0, 0, 0, 0, 1);
    gemm(stream, Eout[0], P(cevw), P(cevb), VE1, Bd * NE, Hd, Hd, Hd, Hd, Hd, 0, 0, 0, 0, 1);
    gemm(stream, Eout[1], P(ceqw), P(ceqb), QE2, Bd * NE, Hd, Hd, Hd, Hd, Hd, 0, 0, 0, 0, 1);
    gemm(stream, Eout[1], P(cekw), P(cekb), KE2, Bd * NE, Hd, Hd, Hd, Hd, Hd, 0, 0, 0, 0, 1);
    gemm(stream, Eout[1], P(cevw), P(cevb), VE2, Bd * NE, Hd, Hd, Hd, Hd, Hd, 0, 0, 0, 0, 1);
    hipLaunchKernelGGL(attn_f32_wmma, dim3(NE / 16, Bd), dim3(32), 0, stream,
                       QE1, KE2, VE2, adj2, cE1, NE, (long long)NE * Hd, (long long)NE * Hd, NE, SCALE);
    hipLaunchKernelGGL(attn_f32_wmma, dim3(NE / 16, Bd), dim3(32), 0, stream,
                       QE2, KE1, VE1, adj1, cE2, NE, (long long)NE * Hd, (long long)NE * Hd, NE, SCALE);

    // ---- edge2node + residual updates ----
    hipLaunchKernelGGL(inc_edge_k, dim3(Bd * Nd), dim3(128), 0, stream, adj1, cE1, t0, Nd);
    gemm(stream, t0, P(e2nw), P(e2nb), t1, Bd * Nd, Hd, Hd, Hd, Hd, Hd, 0, 0, 0, 1, 1);
    hipLaunchKernelGGL(add3_k, dim3((HB + 255) / 256), dim3(256), 0, stream, hb1, ch1, t1, HB);
    hipLaunchKernelGGL(inc_edge_k, dim3(Bd * Nd), dim3(128), 0, stream, adj2, cE2, t0, Nd);
    gemm(stream, t0, P(e2nw), P(e2nb), t1, Bd * Nd, Hd, Hd, Hd, Hd, Hd, 0, 0, 0, 1, 1);
    hipLaunchKernelGGL(add3_k, dim3((HB + 255) / 256), dim3(256), 0, stream, hb2, ch2, t1, HB);
  }

  // ---- readout ----
  hipLaunchKernelGGL(readout_g_k, dim3(Bd), dim3(128), 0, stream, hb1, mf1, gbuf, Nd, 0);
  hipLaunchKernelGGL(readout_g_k, dim3(Bd), dim3(128), 0, stream, hb2, mf2, gbuf, Nd, 128);
  gemm(stream, gbuf, P(ro1w), P(ro1b), r1, Bd, 128, 256, 256, 128, 128, 0, 0, 0, 1, 1);
  gemm(stream, r1, P(ro2w), P(ro2b), (float*)d_out, Bd, 1, 128, 128, 1, 1, 0, 0, 0, 0, 1);
  (void)out_size;
}